// FFTLongConv_2929167696303
// MI455X (gfx1250) — compile-verified
//
#include <hip/hip_runtime.h>
#include <hip/hip_bf16.h>
#include <stdint.h>

// FFT long conv on gfx1250: per-row packed real FFT (z = x + i*h), Stockham
// radix-16 stages executed as 16x16 complex matmuls on V_WMMA_F32_16X16X4_F32,
// async global->LDS loads, 256KB ping-pong LDS, natural-order spectra
// (no bit reversal), hermitian unpack/multiply/repack, inverse FFT, scale 1/N^2.

typedef __attribute__((ext_vector_type(2))) float v2f;
typedef __attribute__((ext_vector_type(8))) float v8f;

#define N_FFT   16384
#define L_SEQ   8192
#define THREADS 256
#define PI_F    3.14159265358979f

__device__ __forceinline__ void async_ld_b128(uint32_t lds_off, const void* g) {
  asm volatile("global_load_async_to_lds_b128 %0, %1, off"
               :: "v"(lds_off), "v"((unsigned long long)(uintptr_t)g)
               : "memory");
}

// One Stockham radix-16 DIF stage. s = 2^ls (stride), m = n/16, n = current size.
// Columns (independent 16-point DFTs): col in [0,1024); q = col & (s-1), p = col >> ls.
// Input element r of a column:  src[q + ((p + m*r)  << ls)]
// Output element k of a column: dst[q + ((16p + k) << ls)] * w_n^{+-pk}
template<bool INV>
__device__ __forceinline__ void radix16_wmma_stage(
    const float* __restrict__ sre, const float* __restrict__ sim,
    float* __restrict__ dre, float* __restrict__ dim_,
    int ls, int m, int n, int wave, int lane,
    const v2f C[4], const v2f S[4], const v2f NS[4])
{
  const int half   = lane >> 4;
  const int col_lo = lane & 15;
  const int smask  = (1 << ls) - 1;
  const float inv_n = 1.0f / (float)n;

  for (int it = 0; it < 8; ++it) {          // 8 waves x 8 tiles = 64 tiles = 1024 cols
    const int tile = wave * 8 + it;
    const int col  = (tile << 4) + col_lo;
    const int q    = col & smask;
    const int p    = col >> ls;

    // Gather B operands (K-chunked 4x16 tiles) straight from LDS.
    v2f br[4], bi[4];
#pragma unroll
    for (int kk = 0; kk < 4; ++kk) {
#pragma unroll
      for (int j = 0; j < 2; ++j) {
        const int r   = 4 * kk + j + 2 * half;
        const int idx = q + ((p + m * r) << ls);
        br[kk][j] = sre[idx];
        bi[kk][j] = sim[idx];
      }
    }

    // Complex 16-pt DFT via f32 WMMA, K accumulated over 4 chunks.
    // fwd: Dr = cos*Br + sin*Bi ; Di = cos*Bi - sin*Br
    // inv: Dr = cos*Br - sin*Bi ; Di = cos*Bi + sin*Br
    v8f dr = {0.f,0.f,0.f,0.f,0.f,0.f,0.f,0.f};
    v8f di = {0.f,0.f,0.f,0.f,0.f,0.f,0.f,0.f};
#pragma unroll
    for (int kk = 0; kk < 4; ++kk) {
      dr = __builtin_amdgcn_wmma_f32_16x16x4_f32(false, C[kk], false, br[kk], (short)0, dr, false, false);
      dr = __builtin_amdgcn_wmma_f32_16x16x4_f32(false, (INV ? NS[kk] : S[kk]), false, bi[kk], (short)0, dr, false, false);
      di = __builtin_amdgcn_wmma_f32_16x16x4_f32(false, C[kk], false, bi[kk], (short)0, di, false, false);
      di = __builtin_amdgcn_wmma_f32_16x16x4_f32(false, (INV ? S[kk] : NS[kk]), false, br[kk], (short)0, di, false, false);
    }

    // Twiddle w_n^{pk} (conjugated for inverse) and scatter from C/D layout.
#pragma unroll
    for (int v = 0; v < 8; ++v) {
      const int k  = v + 8 * half;
      const float th = (INV ? 2.0f : -2.0f) * PI_F * (float)(p * k) * inv_n;
      float sv, cv;
      __sincosf(th, &sv, &cv);
      const float yr = dr[v] * cv - di[v] * sv;
      const float yi = dr[v] * sv + di[v] * cv;
      const int od = q + (((p << 4) + k) << ls);
      dre[od]  = yr;
      dim_[od] = yi;
    }
  }
}

// Final Stockham stage: n=4, s=4096, m=1 (p=0 -> unit twiddles).
template<bool INV>
__device__ __forceinline__ void radix4_stage(
    const float* __restrict__ sre, const float* __restrict__ sim,
    float* __restrict__ dre, float* __restrict__ dim_, int tid)
{
  for (int q = tid; q < 4096; q += THREADS) {
    const float a0r = sre[q],         a0i = sim[q];
    const float a1r = sre[q + 4096],  a1i = sim[q + 4096];
    const float a2r = sre[q + 8192],  a2i = sim[q + 8192];
    const float a3r = sre[q + 12288], a3i = sim[q + 12288];
    const float t0r = a0r + a2r, t0i = a0i + a2i;
    const float t1r = a0r - a2r, t1i = a0i - a2i;
    const float t2r = a1r + a3r, t2i = a1i + a3i;
    const float t3r = a1r - a3r, t3i = a1i - a3i;
    const float br_ = INV ? -t3i : t3i;   // fwd: -i*t3, inv: +i*t3
    const float bi_ = INV ?  t3r : -t3r;
    dre[q]          = t0r + t2r;  dim_[q]          = t0i + t2i;
    dre[q + 4096]   = t1r + br_;  dim_[q + 4096]   = t1i + bi_;
    dre[q + 8192]   = t0r - t2r;  dim_[q + 8192]   = t0i - t2i;
    dre[q + 12288]  = t1r - br_;  dim_[q + 12288]  = t1i - bi_;
  }
}

__global__ void __launch_bounds__(THREADS)
fftconv_kernel(const float* __restrict__ x,
               const float* __restrict__ h,
               float* __restrict__ out)
{
  extern __shared__ float smem[];
  float* Are = smem;
  float* Aim = smem + N_FFT;
  float* Bre = smem + 2 * N_FFT;
  float* Bim = smem + 3 * N_FFT;

  const int tid  = threadIdx.x;
  const int lane = tid & 31;
  const int wave = tid >> 5;
  const int b    = blockIdx.x;

  const float* xrow = x + (size_t)b * L_SEQ;
  const float* hrow = h + (size_t)b * L_SEQ;

  // --- async global -> LDS: pack z = x + i*h, zero-pad to N ---
  {
    const uint32_t are_off = (uint32_t)(uintptr_t)Are;
    const uint32_t aim_off = (uint32_t)(uintptr_t)Aim;
    for (int i = tid; i < L_SEQ / 4; i += THREADS) {
      async_ld_b128(are_off + i * 16, xrow + i * 4);
      async_ld_b128(aim_off + i * 16, hrow + i * 4);
    }
  }
  for (int i = tid; i < L_SEQ; i += THREADS) {
    Are[L_SEQ + i] = 0.f;
    Aim[L_SEQ + i] = 0.f;
  }
  asm volatile("s_wait_asynccnt 0" ::: "memory");
  __syncthreads();

  // --- constant DFT-16 matrix in WMMA A-operand layout ---
  v2f C[4], S[4], NS[4];
  {
    const int mrow  = lane & 15;
    const int khalf = (lane >> 4) * 2;
#pragma unroll
    for (int kk = 0; kk < 4; ++kk) {
#pragma unroll
      for (int j = 0; j < 2; ++j) {
        const int k = 4 * kk + j + khalf;
        const float ang = (2.0f * PI_F / 16.0f) * (float)(mrow * k);
        float sv, cv;
        __sincosf(ang, &sv, &cv);
        C[kk][j] = cv; S[kk][j] = sv; NS[kk][j] = -sv;
      }
    }
  }

  // --- forward FFT (natural order out): 16,16,16,4 ---
  radix16_wmma_stage<false>(Are, Aim, Bre, Bim, 0, 1024, 16384, wave, lane, C, S, NS);
  __syncthreads();
  radix16_wmma_stage<false>(Bre, Bim, Are, Aim, 4, 64, 1024, wave, lane, C, S, NS);
  __syncthreads();
  radix16_wmma_stage<false>(Are, Aim, Bre, Bim, 8, 4, 64, wave, lane, C, S, NS);
  __syncthreads();
  radix4_stage<false>(Bre, Bim, Are, Aim, tid);
  __syncthreads();

  // --- hermitian unpack X,H from packed spectrum, Y = X*H, repack ---
  for (int k = tid; k <= L_SEQ; k += THREADS) {
    const int k2 = (N_FFT - k) & (N_FFT - 1);
    const float ar = Are[k],  ai = Aim[k];
    const float cr = Are[k2], ci = Aim[k2];
    const float xr = 0.5f * (ar + cr), xi = 0.5f * (ai - ci);   // X = (Z + conj(Zr))/2
    const float hr = 0.5f * (ai + ci), hi = 0.5f * (cr - ar);   // H = (Z - conj(Zr))/(2i)
    const float yr = xr * hr - xi * hi;
    const float yi = xr * hi + xi * hr;
    Are[k]  = yr;  Aim[k]  = yi;
    Are[k2] = yr;  Aim[k2] = -yi;   // hermitian symmetry of Y
  }
  __syncthreads();

  // --- inverse FFT (unnormalized) ---
  radix16_wmma_stage<true>(Are, Aim, Bre, Bim, 0, 1024, 16384, wave, lane, C, S, NS);
  __syncthreads();
  radix16_wmma_stage<true>(Bre, Bim, Are, Aim, 4, 64, 1024, wave, lane, C, S, NS);
  __syncthreads();
  radix16_wmma_stage<true>(Are, Aim, Bre, Bim, 8, 4, 64, wave, lane, C, S, NS);
  __syncthreads();
  radix4_stage<true>(Bre, Bim, Are, Aim, tid);
  __syncthreads();

  // --- write first L samples, scale 1/N^2 (matches jnp norm="forward" net 1/n) ---
  const float scale = 1.0f / ((float)N_FFT * (float)N_FFT);
  float* orow = out + (size_t)b * L_SEQ;
  for (int t = tid; t < L_SEQ; t += THREADS) {
    orow[t] = Are[t] * scale;
  }
}

extern "C" void kernel_launch(void* const* d_in, const int* in_sizes, int n_in,
                              void* d_out, int out_size, void* d_ws, size_t ws_size,
                              hipStream_t stream) {
  const float* x = (const float*)d_in[0];
  const float* h = (const float*)d_in[1];
  float* out = (float*)d_out;

  const int batch = in_sizes[0] / L_SEQ;                 // 2048
  const size_t shmem = 4 * (size_t)N_FFT * sizeof(float); // 256 KB dynamic LDS

  hipFuncSetAttribute((const void*)fftconv_kernel,
                      hipFuncAttributeMaxDynamicSharedMemorySize, (int)shmem);

  fftconv_kernel<<<batch, THREADS, shmem, stream>>>(x, h, out);
}